// Transform_38723425141290
// MI455X (gfx1250) — compile-verified
//
#include <hip/hip_runtime.h>
#include <hip/hip_bf16.h>

typedef float v2f __attribute__((ext_vector_type(2)));
typedef float v8f __attribute__((ext_vector_type(8)));

// select among 3 values by index n (0..2), else 0 — lowers to v_cndmask chain
__device__ __forceinline__ float sel3(float a, float b, float c, unsigned n) {
    return (n == 0u) ? a : ((n == 1u) ? b : ((n == 2u) ? c : 0.0f));
}

// R = RX @ RY @ RZ expanded analytically (matches the reference).
__device__ __forceinline__ void make_R(const float* __restrict__ ori,
                                       float R[9]) {
    const float cr = cosf(ori[0]), sr = sinf(ori[0]);
    const float cp = cosf(ori[1]), sp = sinf(ori[1]);
    const float cy = cosf(ori[2]), sy = sinf(ori[2]);
    R[0] = cp * cy;
    R[1] = -cp * sy;
    R[2] = sp;
    R[3] = cr * sy + sr * sp * cy;
    R[4] = cr * cy - sr * sp * sy;
    R[5] = -sr * cp;
    R[6] = sr * sy - cr * sp * cy;
    R[7] = sr * cy + cr * sp * sy;
    R[8] = cr * cp;
}

// Full-tile kernel: each V_WMMA_F32_16X16X4_F32 transforms 16 vertices.
// A (16x4) = 16 x [x y z 1]; B (4x16) = [[R],[t]] zero-padded; D cols 0..2 = out.
// All element offsets fit in 32 bits (tensor is 169 MB).
__global__ void Transform_38723425141290_kernel(const float* __restrict__ J,
                                                const float* __restrict__ ori,
                                                const float* __restrict__ trn,
                                                float* __restrict__ out,
                                                unsigned numFullTiles) {
    float R[9];
    make_R(ori, R);
    const float t0 = trn[0], t1 = trn[1], t2 = trn[2];

    const unsigned lane = threadIdx.x & 31u;
    const unsigned half = lane >> 4;   // 0: lanes 0-15, 1: lanes 16-31
    const unsigned n    = lane & 15u;  // column / vertex-in-tile index

    // B operand (4x16 f32): VGPR0 = row K=(half?2:0), VGPR1 = row K=(half?3:1).
    v2f bmat;
    bmat.x = half ? sel3(R[6], R[7], R[8], n) : sel3(R[0], R[1], R[2], n);
    bmat.y = half ? sel3(t0,   t1,   t2,   n) : sel3(R[3], R[4], R[5], n);

    // Per-lane constant pieces of the tile-relative element offsets.
    const unsigned cx = 3u * n + (half ? 2u : 0u);   // a.x: x or z of vertex n
    const unsigned ch = (half ? 24u : 0u) + n;       // store base: (half*8)*3 + n

    const unsigned wavesTotal = gridDim.x * (blockDim.x >> 5);
    const unsigned waveId =
        (blockIdx.x * blockDim.x + threadIdx.x) >> 5;

    for (unsigned t = waveId; t < numFullTiles; t += wavesTotal) {
        const unsigned e0   = t * 48u;        // dword index of tile start
        const unsigned offx = e0 + cx;
        const unsigned offy = half ? 0u : (offx + 1u);  // dummy J[0] for half=1

        v2f amat;
        amat.x = __builtin_nontemporal_load(J + offx);
        const float ly = __builtin_nontemporal_load(J + offy);
        amat.y = half ? 1.0f : ly;            // K=3 row of A is all-ones

        v8f cacc = {0.f, 0.f, 0.f, 0.f, 0.f, 0.f, 0.f, 0.f};
        // EXEC is all-1s here: loop bound is wave-uniform, selects are cndmask.
        v8f d = __builtin_amdgcn_wmma_f32_16x16x4_f32(
            false, amat, false, bmat, (short)0, cacc, false, false);

        // D (16x16): VGPR r holds vertex v0 + r + half*8, column n. Cols 0..2 used.
        if (n < 3u) {
            const unsigned ob = e0 + ch;
#pragma unroll
            for (int r = 0; r < 8; ++r) {
                __builtin_nontemporal_store(d[r], out + ob + 3u * (unsigned)r);
            }
        }
    }
}

// Scalar tail for numVerts % 16 leftover vertices (not launched when 0).
__global__ void Transform_38723425141290_tail(const float* __restrict__ J,
                                              const float* __restrict__ ori,
                                              const float* __restrict__ trn,
                                              float* __restrict__ out,
                                              unsigned startVert,
                                              unsigned numVerts) {
    const unsigned v = startVert + blockIdx.x * blockDim.x + threadIdx.x;
    if (v >= numVerts) return;
    float R[9];
    make_R(ori, R);
    const float x = J[3u * v + 0u];
    const float y = J[3u * v + 1u];
    const float z = J[3u * v + 2u];
    // joints @ R: out[j] = x*R[0][j] + y*R[1][j] + z*R[2][j] + t[j]
    out[3u * v + 0u] = x * R[0] + y * R[3] + z * R[6] + trn[0];
    out[3u * v + 1u] = x * R[1] + y * R[4] + z * R[7] + trn[1];
    out[3u * v + 2u] = x * R[2] + y * R[5] + z * R[8] + trn[2];
}

extern "C" void kernel_launch(void* const* d_in, const int* in_sizes, int n_in,
                              void* d_out, int out_size, void* d_ws, size_t ws_size,
                              hipStream_t stream) {
    const float* joints      = (const float*)d_in[0];
    const float* orientation = (const float*)d_in[1];
    const float* translation = (const float*)d_in[2];
    float* out = (float*)d_out;

    const unsigned numVerts  = (unsigned)(in_sizes[0] / 3);  // B*N = 14,110,720
    const unsigned fullTiles = numVerts / 16u;
    const unsigned remVerts  = numVerts % 16u;

    if (fullTiles > 0u) {
        const int block = 256;  // 8 waves (wave32)
        unsigned blocksNeeded = (fullTiles + 7u) / 8u;
        if (blocksNeeded < 1u) blocksNeeded = 1u;
        const int grid = (int)((blocksNeeded < 4096u) ? blocksNeeded : 4096u);
        Transform_38723425141290_kernel<<<grid, block, 0, stream>>>(
            joints, orientation, translation, out, fullTiles);
    }
    if (remVerts > 0u) {
        Transform_38723425141290_tail<<<1, 64, 0, stream>>>(
            joints, orientation, translation, out, fullTiles * 16u, numVerts);
    }
}